// MADPSNet_11373073400306
// MI455X (gfx1250) — compile-verified
//
#include <hip/hip_runtime.h>

// CDNA5 / gfx1250, wave32. fp32 matrix path: V_WMMA_F32_16X16X4_F32.
typedef __attribute__((ext_vector_type(2))) float v2f;  // A/B frag: 64 f32 elems / 32 lanes
typedef __attribute__((ext_vector_type(8))) float v8f;  // C/D frag: 16x16 f32 = 256 elems / 32 lanes

#define BM 64          // rows per workgroup tile
#define BN 128         // cols per workgroup tile
#define KT 32          // K slab staged in LDS
#define BSTRIDE (KT + 2)  // padded K-stride for transposed B slab (keeps 8B align, spreads banks)

// C[a] = act( In[a] (MxK) * W[sel[a]] (KxN) + bias[sel[a]] ), row-major in global memory.
// Block = 256 threads = 8 waves; wave (wr,wc) owns a 32x32 patch = 2x2 WMMA tiles.
__global__ __launch_bounds__(256)
void moe_gemm_bias_act(const float* __restrict__ in,    // [A, M, K]
                       const float* __restrict__ W,     // [E, K, N]
                       const float* __restrict__ bias,  // [E, N]
                       const int*   __restrict__ sel,   // [A]  (jax x64-off -> int32)
                       float* __restrict__ out,         // [A, M, N]
                       int M, int N, int K, int relu)
{
    __shared__ float As[BM * KT];        // 8 KB, row-major [m][k]
    __shared__ float Bs[BN * BSTRIDE];   // ~17 KB, transposed [n][k] (padded)

    const int a = blockIdx.z;
    const int e = sel[a];
    const float* inA  = in   + (size_t)a * M * K;
    const float* We   = W    + (size_t)e * K * N;
    const float* be   = bias + (size_t)e * N;
    float*       outA = out  + (size_t)a * M * N;

    const int rowBase = blockIdx.x * BM;
    const int colBase = blockIdx.y * BN;

    const int tid   = threadIdx.x;
    const int lane  = tid & 31;
    const int wave  = tid >> 5;
    const int wr    = (wave & 1) * 32;   // wave row offset within tile
    const int wc    = (wave >> 1) * 32;  // wave col offset within tile

    const int lm    = lane & 15;         // M (A-frag) / N (B-frag) index within 16
    const int khalf = lane >> 4;         // lane half selects K pair {0,1} vs {2,3}

    v8f acc[2][2];
    {
        v8f z = {0.f, 0.f, 0.f, 0.f, 0.f, 0.f, 0.f, 0.f};
        acc[0][0] = z; acc[0][1] = z; acc[1][0] = z; acc[1][1] = z;
    }

    for (int k0 = 0; k0 < K; k0 += KT) {
        // ---- stage A slab: BM x KT = 512 float4, 2 per thread, row-major ----
        #pragma unroll
        for (int i = 0; i < 2; ++i) {
            int idx = tid + i * 256;          // 0..511
            int r   = idx >> 3;               // KT/4 = 8 float4 per row
            int c4  = idx & 7;
            float4 v = *reinterpret_cast<const float4*>(
                inA + (size_t)(rowBase + r) * K + k0 + c4 * 4);
            *reinterpret_cast<float4*>(&As[r * KT + c4 * 4]) = v;
        }
        // ---- stage B slab transposed: global row-major [k][n] -> LDS [n][k] ----
        #pragma unroll
        for (int i = 0; i < 4; ++i) {
            int idx = tid + i * 256;          // 0..1023
            int r   = idx >> 5;               // k within slab, 0..31
            int c4  = idx & 31;               // n group of 4
            float4 v = *reinterpret_cast<const float4*>(
                We + (size_t)(k0 + r) * N + colBase + c4 * 4);
            int nb = c4 * 4;
            Bs[(nb + 0) * BSTRIDE + r] = v.x;
            Bs[(nb + 1) * BSTRIDE + r] = v.y;
            Bs[(nb + 2) * BSTRIDE + r] = v.z;
            Bs[(nb + 3) * BSTRIDE + r] = v.w;
        }
        __syncthreads();

        // ---- compute: 8 K-steps of 4, 2x2 WMMA tiles per wave ----
        // Both fragments are now a single contiguous 8B LDS read per lane ->
        // ds_load_2addr_b64 straight into even-aligned WMMA source pairs.
        #pragma unroll
        for (int kk = 0; kk < KT; kk += 4) {
            v2f afrag[2], bfrag[2];
            #pragma unroll
            for (int rs = 0; rs < 2; ++rs) {
                int m = wr + rs * 16 + lm;
                afrag[rs] = *reinterpret_cast<const v2f*>(&As[m * KT + kk + khalf * 2]);
            }
            #pragma unroll
            for (int cs = 0; cs < 2; ++cs) {
                int n = wc + cs * 16 + lm;
                bfrag[cs] = *reinterpret_cast<const v2f*>(&Bs[n * BSTRIDE + kk + khalf * 2]);
            }
            #pragma unroll
            for (int rs = 0; rs < 2; ++rs)
                #pragma unroll
                for (int cs = 0; cs < 2; ++cs)
                    acc[rs][cs] = __builtin_amdgcn_wmma_f32_16x16x4_f32(
                        /*neg_a=*/false, afrag[rs],
                        /*neg_b=*/false, bfrag[cs],
                        /*c_mod=*/(short)0, acc[rs][cs],
                        /*reuse_a=*/false, /*reuse_b=*/false);
        }
        __syncthreads();
    }

    // ---- epilogue: C/D layout -> VGPR v holds M=v (lanes 0-15) / M=v+8 (lanes 16-31), N=lm ----
    const int mHalf = khalf * 8;
    #pragma unroll
    for (int rs = 0; rs < 2; ++rs) {
        #pragma unroll
        for (int cs = 0; cs < 2; ++cs) {
            int   col = colBase + wc + cs * 16 + lm;
            float bv  = be[col];
            #pragma unroll
            for (int v = 0; v < 8; ++v) {
                int   row = rowBase + wr + rs * 16 + mHalf + v;
                float x   = acc[rs][cs][v] + bv;
                if (relu) x = fmaxf(x, 0.f);
                outA[(size_t)row * N + col] = x;
            }
        }
    }
}

extern "C" void kernel_launch(void* const* d_in, const int* in_sizes, int n_in,
                              void* d_out, int out_size, void* d_ws, size_t ws_size,
                              hipStream_t stream) {
    (void)in_sizes; (void)n_in; (void)out_size; (void)ws_size;

    const float* inputs = (const float*)d_in[0];   // [A,B,S]
    const int*   sel_s  = (const int*)  d_in[1];   // [1,A] laac_shallow
    const int*   sel_d  = (const int*)  d_in[2];   // [1,A] laac_deep
    const float* Ws1    = (const float*)d_in[3];   // [E,S,H1]
    const float* bs1    = (const float*)d_in[4];   // [E,H1]
    const float* Ws2    = (const float*)d_in[5];   // [E,H1,H2]
    const float* bs2    = (const float*)d_in[6];   // [E,H2]
    const float* Wd1    = (const float*)d_in[7];   // [E,D0,D1]
    const float* bd1    = (const float*)d_in[8];   // [E,D1]
    const float* Wd2    = (const float*)d_in[9];   // [E,D1,D2]
    const float* bd2    = (const float*)d_in[10];  // [E,D2]
    float* out = (float*)d_out;                    // [A,B,D2]

    const int A = 8, B = 2048, S = 256, H1 = 512, H2 = 256, D1 = 512, D2 = 128;

    // workspace: h [A,B,H1] 32MB | sh [A,B,H2] 16MB | d [A,B,D1] 32MB = 80MB
    float* h  = (float*)d_ws;
    float* sh = h  + (size_t)A * B * H1;
    float* dd = sh + (size_t)A * B * H2;

    dim3 blk(256);
    // shallow: Linear+ReLU, Linear+ReLU — only the selected shallow expert per agent
    moe_gemm_bias_act<<<dim3(B / BM, H1 / BN, A), blk, 0, stream>>>(
        inputs, Ws1, bs1, sel_s, h,  B, H1, S,  1);
    moe_gemm_bias_act<<<dim3(B / BM, H2 / BN, A), blk, 0, stream>>>(
        h,      Ws2, bs2, sel_s, sh, B, H2, H1, 1);
    // deep: Linear+ReLU, Linear (no final activation) — selected deep expert per agent
    moe_gemm_bias_act<<<dim3(B / BM, D1 / BN, A), blk, 0, stream>>>(
        sh,     Wd1, bd1, sel_d, dd, B, D1, H2, 1);
    moe_gemm_bias_act<<<dim3(B / BM, D2 / BN, A), blk, 0, stream>>>(
        dd,     Wd2, bd2, sel_d, out, B, D2, D1, 0);
}